// Net_87608742903968
// MI455X (gfx1250) — compile-verified
//
#include <hip/hip_runtime.h>
#include <math.h>

#define N_NODES  10000
#define N_EDGES  160000
#define N_GRAPHS 4

typedef __attribute__((ext_vector_type(2))) float v2f;
typedef __attribute__((ext_vector_type(8))) float v8f;

#define INV_SQRT6   0.40824829046386301637f
#define INV_SQRT32  0.17677669529663688110f
#define INV_SQRT96  0.10206207261596575427f
#define INV_SQRT128 0.08838834764831844055f

__device__ __forceinline__ float sigmoidf_(float x) { return 1.f / (1.f + __expf(-x)); }
__device__ __forceinline__ float siluf_(float x)    { return x * sigmoidf_(x); }

__device__ __forceinline__ void atomicAddF(float* p, float v) {
    __hip_atomic_fetch_add(p, v, __ATOMIC_RELAXED, __HIP_MEMORY_SCOPE_AGENT);
}

__device__ __forceinline__ v8f wmma4(v2f a, v2f b, v8f c) {
    // D = A(16x4 f32) * B(4x16 f32) + C  -> v_wmma_f32_16x16x4_f32
    return __builtin_amdgcn_wmma_f32_16x16x4_f32(false, a, false, b, (short)0, c,
                                                 false, false);
}

// ---------------------------------------------------------------------------
// Weight packing: fold e3linear-pre + all path normalizations into combined
// TP weight matrices, stored in WMMA B-fragment order.
// Packed layout per branch (floats):
//   [0,36864)      Ws  288x128  (cols: out0[64] | gate[32] | t[32])
//   [36864,39936)  Wv  96x32
//   [39936,48128)  W1  128x64 (mlp)
//   [48128,52224)  W2  64x64
//   [52224,58368)  W3  64x96
// ---------------------------------------------------------------------------
struct BranchW {
    const float* tp[18];
    const float* preS;   // 64x64 lin_pre scalar (fold) or null
    const float* preV;   // 32x32 lin_pre vector (fold) or null
    const float* w1;
    const float* w2;
    const float* w3;
    int fold;
};

#define PK_WS 0
#define PK_WV 36864
#define PK_W1 39936
#define PK_W2 48128
#define PK_W3 52224
#define PK_TOTAL 58368

__device__ float ws_val(int k, int n, const BranchW& W) {
    int sel, wn;
    if (n < 64)      { sel = 0; wn = n; }
    else if (n < 96) { sel = 1; wn = n - 64; }
    else             { sel = 2; wn = n - 96; }
    const int ncol = (sel == 0) ? 64 : 32;
    const float scale_s = 0.125f * INV_SQRT6;          // 1/sqrt(64) / sqrt(6)
    const float scale_d = INV_SQRT96 * INV_SQRT6;      // CG 1/sqrt3 * 1/sqrt32 / sqrt6
    if (k < 192) {                                     // scalar feature rows
        int blk = k >> 6;                              // 0:s_i 1:s_j 2:s_e
        int u = k & 63;
        const float* P = W.tp[blk * 6 + sel];
        if (blk == 2 && W.fold) {                      // fold lin_pre scalar (1/8)
            float acc = 0.f;
            for (int u2 = 0; u2 < 64; ++u2) acc += W.preS[u * 64 + u2] * P[u2 * ncol + wn];
            return acc * 0.125f * scale_s;
        }
        return P[u * ncol + wn] * scale_s;
    }
    if (sel == 2) return 0.f;                          // dot rows don't feed t
    int blk = (k - 192) >> 5;                          // 0:d_i 1:d_j 2:d_e
    int u = (k - 192) & 31;
    const float* P = W.tp[blk * 6 + 4 + sel];
    if (blk == 2 && W.fold) {                          // fold lin_pre vector (1/sqrt32)
        float acc = 0.f;
        for (int u2 = 0; u2 < 32; ++u2) acc += W.preV[u * 32 + u2] * P[u2 * ncol + wn];
        return acc * INV_SQRT32 * scale_d;
    }
    return P[u * ncol + wn] * scale_d;
}

__device__ float wv_val(int k, int n, const BranchW& W) {
    const float scale_v = INV_SQRT32 * INV_SQRT6;      // vec x sh_s path
    int blk = k >> 5, u = k & 31;
    const float* P = W.tp[blk * 6 + 3];
    if (blk == 2 && W.fold) {
        float acc = 0.f;
        for (int u2 = 0; u2 < 32; ++u2) acc += W.preV[u * 32 + u2] * P[u2 * 32 + n];
        return acc * INV_SQRT32 * scale_v;
    }
    return P[u * 32 + n] * scale_v;
}

__global__ void pack_weights_kernel(BranchW W, float* __restrict__ out) {
    for (int idx = blockIdx.x * blockDim.x + threadIdx.x; idx < PK_TOTAL;
         idx += gridDim.x * blockDim.x) {
        int t, frag, r, lane, j, kb, nt, k, n;
        float val;
        if (idx < PK_WV) {
            t = idx; frag = t >> 6; r = t & 63; lane = r >> 1; j = r & 1;
            kb = frag >> 3; nt = frag & 7;
            k = kb * 4 + 2 * (lane >> 4) + j; n = nt * 16 + (lane & 15);
            val = ws_val(k, n, W);
        } else if (idx < PK_W1) {
            t = idx - PK_WV; frag = t >> 6; r = t & 63; lane = r >> 1; j = r & 1;
            kb = frag >> 1; nt = frag & 1;
            k = kb * 4 + 2 * (lane >> 4) + j; n = nt * 16 + (lane & 15);
            val = wv_val(k, n, W);
        } else if (idx < PK_W2) {
            t = idx - PK_W1; frag = t >> 6; r = t & 63; lane = r >> 1; j = r & 1;
            kb = frag >> 2; nt = frag & 3;
            k = kb * 4 + 2 * (lane >> 4) + j; n = nt * 16 + (lane & 15);
            val = W.w1[k * 64 + n];
        } else if (idx < PK_W3) {
            t = idx - PK_W2; frag = t >> 6; r = t & 63; lane = r >> 1; j = r & 1;
            kb = frag >> 2; nt = frag & 3;
            k = kb * 4 + 2 * (lane >> 4) + j; n = nt * 16 + (lane & 15);
            val = W.w2[k * 64 + n];
        } else {
            t = idx - PK_W3; frag = t >> 6; r = t & 63; lane = r >> 1; j = r & 1;
            kb = frag / 6; nt = frag % 6;
            k = kb * 4 + 2 * (lane >> 4) + j; n = nt * 16 + (lane & 15);
            val = W.w3[k * 96 + n];
        }
        out[idx] = val;
    }
}

// ---------------------------------------------------------------------------
// Mega edge kernel: one wave handles 16 edges.
//   GEMM1 Xs[16x288] @ Ws[288x128] -> [out0|gate|t]
//   GEMM2 Xv[48x96]  @ Wv[96x32]   -> per-component vector output
//   MLP   len_emb[16x128] -> 64 -> 64 -> 96 (silu, silu, lin)
//   gate + cfconv -> z[16x160]
// LDS strides chosen odd-mod-64 patterns for conflict-free WMMA A loads.
// ---------------------------------------------------------------------------
#define XS_S 292
#define XV_S 100
#define SB_S 132
#define VB_S 36
#define H_S  68
#define HM_S 100
// offsets (floats) in shared pool
#define O_XS  0
#define O_XV  4672              // 16*292
#define O_SB  9472              // +48*100
#define O_VB  11584             // +16*132
#define O_SHB 13312             // +48*36
#define SMEM_FLOATS 13376
// MLP buffers overlay the Xv region once GEMM2 is done
#define O_H1  O_XV
#define O_H2  (O_XV + 1088)
#define O_HM  (O_XV + 2176)

__global__ __launch_bounds__(32) void mega_edge_kernel(
    const float* __restrict__ nodeTab, const float* __restrict__ edgeFea,
    const float* __restrict__ sh, const float* __restrict__ lenEmb,
    const int* __restrict__ idxI, const int* __restrict__ idxJ,
    const float* __restrict__ pack,
    const float* __restrict__ b1, const float* __restrict__ b2,
    const float* __restrict__ b3, float* __restrict__ zOut) {
    __shared__ float smem[SMEM_FLOATS];
    float* Xs = smem + O_XS;
    float* Xv = smem + O_XV;
    float* Sb = smem + O_SB;
    float* Vb = smem + O_VB;
    float* shb = smem + O_SHB;
    float* H1 = smem + O_H1;
    float* H2 = smem + O_H2;
    float* Hm = smem + O_HM;

    const int lane = threadIdx.x;
    const int e0 = blockIdx.x * 16;
    const float* WsP = pack + PK_WS;
    const float* WvP = pack + PK_WV;
    const float* W1P = pack + PK_W1;
    const float* W2P = pack + PK_W2;
    const float* W3P = pack + PK_W3;

    for (int e = 0; e < 16; ++e) {
        const int ge = e0 + e;
        const float* A = nodeTab + (long)idxI[ge] * 160;
        const float* B = nodeTab + (long)idxJ[ge] * 160;
        const float* C = edgeFea + (long)ge * 160;
        if (lane < 4) shb[e * 4 + lane] = sh[(long)ge * 4 + lane];
        for (int t = lane; t < 64; t += 32) {
            Xs[e * XS_S + t]       = A[t];
            Xs[e * XS_S + 64 + t]  = B[t];
            Xs[e * XS_S + 128 + t] = C[t];
        }
        const float s1 = sh[(long)ge * 4 + 1];
        const float s2 = sh[(long)ge * 4 + 2];
        const float s3 = sh[(long)ge * 4 + 3];
        const int u = lane;   // 32 lanes = 32 vector channels
        float ax = A[64 + 3 * u], ay = A[65 + 3 * u], az = A[66 + 3 * u];
        Xs[e * XS_S + 192 + u] = ax * s1 + ay * s2 + az * s3;
        Xv[(3 * e + 0) * XV_S + u] = ax;
        Xv[(3 * e + 1) * XV_S + u] = ay;
        Xv[(3 * e + 2) * XV_S + u] = az;
        float bx = B[64 + 3 * u], by = B[65 + 3 * u], bz = B[66 + 3 * u];
        Xs[e * XS_S + 224 + u] = bx * s1 + by * s2 + bz * s3;
        Xv[(3 * e + 0) * XV_S + 32 + u] = bx;
        Xv[(3 * e + 1) * XV_S + 32 + u] = by;
        Xv[(3 * e + 2) * XV_S + 32 + u] = bz;
        float cx = C[64 + 3 * u], cy = C[65 + 3 * u], cz = C[66 + 3 * u];
        Xs[e * XS_S + 256 + u] = cx * s1 + cy * s2 + cz * s3;
        Xv[(3 * e + 0) * XV_S + 64 + u] = cx;
        Xv[(3 * e + 1) * XV_S + 64 + u] = cy;
        Xv[(3 * e + 2) * XV_S + 64 + u] = cz;
    }
    __syncthreads();

    const int m = lane & 15;
    const int kp = (lane >> 4) * 2;
    const v8f vzero = {0.f, 0.f, 0.f, 0.f, 0.f, 0.f, 0.f, 0.f};

    // GEMM1: 288-K, 8 n-tiles (pairs for ILP)
    for (int nt = 0; nt < 8; nt += 2) {
        v8f acc0 = vzero, acc1 = vzero;
        for (int kb = 0; kb < 72; ++kb) {
            v2f a;
            a.x = Xs[m * XS_S + kb * 4 + kp];
            a.y = Xs[m * XS_S + kb * 4 + kp + 1];
            v2f bA = *(const v2f*)(WsP + ((kb * 8 + nt) * 64 + lane * 2));
            v2f bB = *(const v2f*)(WsP + ((kb * 8 + nt + 1) * 64 + lane * 2));
            acc0 = wmma4(a, bA, acc0);
            acc1 = wmma4(a, bB, acc1);
        }
        for (int v = 0; v < 8; ++v) {
            int row = v + 8 * (lane >> 4);
            Sb[row * SB_S + nt * 16 + m] = acc0[v];
            Sb[row * SB_S + (nt + 1) * 16 + m] = acc1[v];
        }
    }
    // GEMM2: 48 rows (16 edges x 3 comps), 96-K, 2 n-tiles
    for (int mt = 0; mt < 3; ++mt) {
        v8f acc0 = vzero, acc1 = vzero;
        for (int kb = 0; kb < 24; ++kb) {
            v2f a;
            a.x = Xv[(mt * 16 + m) * XV_S + kb * 4 + kp];
            a.y = Xv[(mt * 16 + m) * XV_S + kb * 4 + kp + 1];
            v2f bA = *(const v2f*)(WvP + ((kb * 2 + 0) * 64 + lane * 2));
            v2f bB = *(const v2f*)(WvP + ((kb * 2 + 1) * 64 + lane * 2));
            acc0 = wmma4(a, bA, acc0);
            acc1 = wmma4(a, bB, acc1);
        }
        for (int v = 0; v < 8; ++v) {
            int row = mt * 16 + v + 8 * (lane >> 4);
            Vb[row * VB_S + m] = acc0[v];
            Vb[row * VB_S + 16 + m] = acc1[v];
        }
    }
    __syncthreads();

    // MLP input overlays Xs (free after GEMM1)
    for (int e = 0; e < 16; ++e)
        for (int t = lane; t < 128; t += 32)
            Xs[e * XS_S + t] = lenEmb[(long)(e0 + e) * 128 + t];
    __syncthreads();

    // MLP layer 1: 128 -> 64, silu
    for (int nt = 0; nt < 4; nt += 2) {
        v8f acc0 = vzero, acc1 = vzero;
        for (int kb = 0; kb < 32; ++kb) {
            v2f a;
            a.x = Xs[m * XS_S + kb * 4 + kp];
            a.y = Xs[m * XS_S + kb * 4 + kp + 1];
            v2f bA = *(const v2f*)(W1P + ((kb * 4 + nt) * 64 + lane * 2));
            v2f bB = *(const v2f*)(W1P + ((kb * 4 + nt + 1) * 64 + lane * 2));
            acc0 = wmma4(a, bA, acc0);
            acc1 = wmma4(a, bB, acc1);
        }
        for (int v = 0; v < 8; ++v) {
            int row = v + 8 * (lane >> 4);
            int c0 = nt * 16 + m, c1 = (nt + 1) * 16 + m;
            H1[row * H_S + c0] = siluf_(acc0[v] + b1[c0]);
            H1[row * H_S + c1] = siluf_(acc1[v] + b1[c1]);
        }
    }
    __syncthreads();
    // MLP layer 2: 64 -> 64, silu
    for (int nt = 0; nt < 4; nt += 2) {
        v8f acc0 = vzero, acc1 = vzero;
        for (int kb = 0; kb < 16; ++kb) {
            v2f a;
            a.x = H1[m * H_S + kb * 4 + kp];
            a.y = H1[m * H_S + kb * 4 + kp + 1];
            v2f bA = *(const v2f*)(W2P + ((kb * 4 + nt) * 64 + lane * 2));
            v2f bB = *(const v2f*)(W2P + ((kb * 4 + nt + 1) * 64 + lane * 2));
            acc0 = wmma4(a, bA, acc0);
            acc1 = wmma4(a, bB, acc1);
        }
        for (int v = 0; v < 8; ++v) {
            int row = v + 8 * (lane >> 4);
            int c0 = nt * 16 + m, c1 = (nt + 1) * 16 + m;
            H2[row * H_S + c0] = siluf_(acc0[v] + b2[c0]);
            H2[row * H_S + c1] = siluf_(acc1[v] + b2[c1]);
        }
    }
    __syncthreads();
    // MLP layer 3: 64 -> 96
    for (int nt = 0; nt < 6; nt += 2) {
        v8f acc0 = vzero, acc1 = vzero;
        for (int kb = 0; kb < 16; ++kb) {
            v2f a;
            a.x = H2[m * H_S + kb * 4 + kp];
            a.y = H2[m * H_S + kb * 4 + kp + 1];
            v2f bA = *(const v2f*)(W3P + ((kb * 6 + nt) * 64 + lane * 2));
            v2f bB = *(const v2f*)(W3P + ((kb * 6 + nt + 1) * 64 + lane * 2));
            acc0 = wmma4(a, bA, acc0);
            acc1 = wmma4(a, bB, acc1);
        }
        for (int v = 0; v < 8; ++v) {
            int row = v + 8 * (lane >> 4);
            int c0 = nt * 16 + m, c1 = (nt + 1) * 16 + m;
            Hm[row * HM_S + c0] = acc0[v] + b3[c0];
            Hm[row * HM_S + c1] = acc1[v] + b3[c1];
        }
    }
    __syncthreads();

    // gate + cfconv, write z[16x160]
    for (int t = lane; t < 16 * 64; t += 32) {
        int e = t >> 6, w = t & 63;
        float sv = Sb[e * SB_S + w];
        zOut[(long)(e0 + e) * 160 + w] = siluf_(sv) * Hm[e * HM_S + w];
    }
    for (int t = lane; t < 16 * 32; t += 32) {
        int e = t >> 5, w = t & 31;
        float g  = sigmoidf_(Sb[e * SB_S + 64 + w]);
        float tt = Sb[e * SB_S + 96 + w];
        float gw = g * Hm[e * HM_S + 64 + w];
        float s0 = shb[e * 4 + 0];
        long base = (long)(e0 + e) * 160 + 64 + 3 * w;
        zOut[base + 0] = gw * (tt * shb[e * 4 + 1] + Vb[(3 * e + 0) * VB_S + w] * s0);
        zOut[base + 1] = gw * (tt * shb[e * 4 + 2] + Vb[(3 * e + 1) * VB_S + w] * s0);
        zOut[base + 2] = gw * (tt * shb[e * 4 + 3] + Vb[(3 * e + 2) * VB_S + w] * s0);
    }
}

// ---------------------------------------------------------------------------
// Small per-row kernels (node side is 16x cheaper than edges; keep VALU)
// ---------------------------------------------------------------------------
__global__ void zero_kernel(float* __restrict__ agg, float* __restrict__ stN,
                            float* __restrict__ stE) {
    int i = blockIdx.x * blockDim.x + threadIdx.x;
    if (i < N_NODES * 160) agg[i] = 0.f;
    if (i < 16) { stN[i] = 0.f; stE[i] = 0.f; }
}

__global__ void node_pre_kernel(const float* __restrict__ X,
                                const float* __restrict__ Ws,
                                const float* __restrict__ Wv,
                                float* __restrict__ Y) {
    __shared__ float row[160];
    int n = blockIdx.x, t = threadIdx.x;
    row[t] = X[(long)n * 160 + t];
    __syncthreads();
    float acc = 0.f;
    if (t < 64) {
        for (int u = 0; u < 64; ++u) acc += row[u] * Ws[u * 64 + t];
        Y[(long)n * 160 + t] = acc * 0.125f;
    } else {
        int w = (t - 64) / 3, c = (t - 64) % 3;
        for (int u = 0; u < 32; ++u) acc += row[64 + 3 * u + c] * Wv[u * 32 + w];
        Y[(long)n * 160 + t] = acc * INV_SQRT32;
    }
}

__global__ void scatter_add_kernel(const float* __restrict__ z,
                                   const int* __restrict__ idxI,
                                   float* __restrict__ agg) {
    int e = blockIdx.x, t = threadIdx.x;
    float v = z[(long)e * 160 + t] * 0.25f;   // / sqrt(E/N) = 1/4
    atomicAddF(&agg[(long)idxI[e] * 160 + t], v);
}

__global__ void node_post_kernel(const float* __restrict__ agg,
                                 const float* __restrict__ fea,
                                 const float* __restrict__ oh,
                                 const float* __restrict__ Wps,
                                 const float* __restrict__ Wpv,
                                 const float* __restrict__ Wsc0,
                                 const float* __restrict__ Wsc1,
                                 float* __restrict__ nf2) {
    __shared__ float ar[160], fr[160];
    int n = blockIdx.x, t = threadIdx.x;
    ar[t] = agg[(long)n * 160 + t];
    fr[t] = fea[(long)n * 160 + t];
    __syncthreads();
    int s = (oh[(long)n * 2 + 1] > 0.5f) ? 1 : 0;
    float acc = 0.f, slf = 0.f;
    if (t < 64) {
        for (int u = 0; u < 64; ++u) {
            acc += ar[u] * Wps[u * 64 + t];
            slf += fr[u] * Wsc0[(u * 2 + s) * 64 + t];
        }
        nf2[(long)n * 160 + t] = acc * 0.125f + slf * INV_SQRT128;
    } else {
        int w = (t - 64) / 3, c = (t - 64) % 3;
        for (int u = 0; u < 32; ++u) {
            acc += ar[64 + 3 * u + c] * Wpv[u * 32 + w];
            slf += fr[64 + 3 * u + c] * Wsc1[(u * 2 + s) * 32 + w];
        }
        nf2[(long)n * 160 + t] = acc * INV_SQRT32 + slf * 0.125f;
    }
}

__global__ void edge_post_kernel(float* __restrict__ z,
                                 const float* __restrict__ fea,
                                 const float* __restrict__ oh4,
                                 const float* __restrict__ Wps,
                                 const float* __restrict__ Wpv,
                                 const float* __restrict__ Wsc0,
                                 const float* __restrict__ Wsc1) {
    __shared__ float zr[160], fr[160];
    int e = blockIdx.x, t = threadIdx.x;
    zr[t] = z[(long)e * 160 + t];
    fr[t] = fea[(long)e * 160 + t];
    __syncthreads();
    const float* o = oh4 + (long)e * 4;
    int p = (o[1] > 0.5f) ? 1 : ((o[2] > 0.5f) ? 2 : ((o[3] > 0.5f) ? 3 : 0));
    float acc = 0.f, slf = 0.f;
    if (t < 64) {
        for (int u = 0; u < 64; ++u) {
            acc += zr[u] * Wps[u * 64 + t];
            slf += fr[u] * Wsc0[(u * 4 + p) * 64 + t];
        }
        z[(long)e * 160 + t] = acc * 0.125f + slf * (1.f / 16.f);
    } else {
        int w = (t - 64) / 3, c = (t - 64) % 3;
        for (int u = 0; u < 32; ++u) {
            acc += zr[64 + 3 * u + c] * Wpv[u * 32 + w];
            slf += fr[64 + 3 * u + c] * Wsc1[(u * 4 + p) * 32 + w];
        }
        z[(long)e * 160 + t] = acc * INV_SQRT32 + slf * INV_SQRT128;
    }
}

// stats layout: [0..3]=cnt, [4..7]=scalar-sum, [8..11]=scalar-sqsum, [12..15]=vec-sqsum
__global__ void ln_stats1_kernel(const float* __restrict__ Z,
                                 const int* __restrict__ batch,
                                 const int* __restrict__ gidx, int rows,
                                 float* __restrict__ st) {
    int r = blockIdx.x * blockDim.x + threadIdx.x;
    if (r >= rows) return;
    int g = batch[gidx ? gidx[r] : r];
    float s = 0.f;
    for (int u = 0; u < 64; ++u) s += Z[(long)r * 160 + u];
    atomicAddF(&st[g], 1.f);
    atomicAddF(&st[4 + g], s);
}

__global__ void ln_stats2_kernel(const float* __restrict__ Z,
                                 const int* __restrict__ batch,
                                 const int* __restrict__ gidx, int rows,
                                 float* __restrict__ st) {
    int r = blockIdx.x * blockDim.x + threadIdx.x;
    if (r >= rows) return;
    int g = batch[gidx ? gidx[r] : r];
    float cnt = fmaxf(st[g], 1.f);
    float mean = st[4 + g] / (cnt * 64.f);
    float ss = 0.f, sv = 0.f;
    for (int u = 0; u < 64; ++u) {
        float d = Z[(long)r * 160 + u] - mean;
        ss += d * d;
    }
    for (int t = 64; t < 160; ++t) {
        float v = Z[(long)r * 160 + t];
        sv += v * v;
    }
    atomicAddF(&st[8 + g], ss);
    atomicAddF(&st[12 + g], sv);
}

__global__ void ln_finish_kernel(const float* __restrict__ Z,
                                 const float* __restrict__ base,
                                 const int* __restrict__ batch,
                                 const int* __restrict__ gidx,
                                 const float* __restrict__ gamma,
                                 const float* __restrict__ beta,
                                 const float* __restrict__ st,
                                 float* __restrict__ out) {
    int r = blockIdx.x, t = threadIdx.x;
    int g = batch[gidx ? gidx[r] : r];
    float cnt = fmaxf(st[g], 1.f);
    float mean = st[4 + g] / (cnt * 64.f);
    float rs = rsqrtf(st[8 + g] / (cnt * 64.f) + 1e-5f);
    float rv = rsqrtf(st[12 + g] / (cnt * 96.f) + 1e-5f);
    float zv = Z[(long)r * 160 + t];
    float val;
    if (t < 64) val = (zv - mean) * rs * gamma[t] + beta[t];
    else { int w = (t - 64) / 3; val = zv * rv * gamma[64 + w]; }
    out[(long)r * 160 + t] = base[(long)r * 160 + t] + val;
}

// ---------------------------------------------------------------------------
extern "C" void kernel_launch(void* const* d_in, const int* in_sizes, int n_in,
                              void* d_out, int out_size, void* d_ws, size_t ws_size,
                              hipStream_t stream) {
    (void)in_sizes; (void)n_in; (void)out_size; (void)ws_size;
    const float* node_fea = (const float*)d_in[0];
    const float* edge_sh  = (const float*)d_in[1];
    const float* edge_fea = (const float*)d_in[2];
    const float* len_emb  = (const float*)d_in[3];
    const float* node_oh  = (const float*)d_in[4];
    const float* edge_oh  = (const float*)d_in[5];
    const int*   eidx     = (const int*)d_in[6];
    const int*   batch    = (const int*)d_in[7];
    const int* idxI = eidx;
    const int* idxJ = eidx + N_EDGES;

    // params in dict-insertion order
    const float* n_sc0  = (const float*)d_in[8];
    const float* n_sc1  = (const float*)d_in[9];
    const float* n_pre0 = (const float*)d_in[10];
    const float* n_pre1 = (const float*)d_in[11];
    // n_tp = 12..29
    const float* n_b1 = (const float*)d_in[31];
    const float* n_b2 = (const float*)d_in[33];
    const float* n_b3 = (const float*)d_in[35];
    const float* n_post0 = (const float*)d_in[36];
    const float* n_post1 = (const float*)d_in[37];
    const float* n_gamma = (const float*)d_in[38];
    const float* n_beta  = (const float*)d_in[39];
    const float* e_sc0  = (const float*)d_in[40];
    const float* e_sc1  = (const float*)d_in[41];
    const float* e_pre0 = (const float*)d_in[42];
    const float* e_pre1 = (const float*)d_in[43];
    // e_tp = 44..61
    const float* e_b1 = (const float*)d_in[63];
    const float* e_b2 = (const float*)d_in[65];
    const float* e_b3 = (const float*)d_in[67];
    const float* e_post0 = (const float*)d_in[68];
    const float* e_post1 = (const float*)d_in[69];
    const float* e_gamma = (const float*)d_in[70];
    const float* e_beta  = (const float*)d_in[71];

    float* base    = (float*)d_ws;
    float* nf      = base;                     // 1.6M
    float* agg     = base + 1600000;           // 1.6M
    float* nf2     = base + 3200000;           // 1.6M
    float* nodeOut = base + 4800000;           // 1.6M
    float* zbuf    = base + 6400000;           // 25.6M
    float* packN   = base + 32000000;          // 58368
    float* packE   = base + 32058368;          // 58368
    float* stN     = base + 32116736;          // 16
    float* stE     = base + 32116752;          // 16

    BranchW WN;
    for (int t = 0; t < 18; ++t) WN.tp[t] = (const float*)d_in[12 + t];
    WN.preS = nullptr; WN.preV = nullptr;
    WN.w1 = (const float*)d_in[30];
    WN.w2 = (const float*)d_in[32];
    WN.w3 = (const float*)d_in[34];
    WN.fold = 0;

    BranchW WE;
    for (int t = 0; t < 18; ++t) WE.tp[t] = (const float*)d_in[44 + t];
    WE.preS = e_pre0; WE.preV = e_pre1;
    WE.w1 = (const float*)d_in[62];
    WE.w2 = (const float*)d_in[64];
    WE.w3 = (const float*)d_in[66];
    WE.fold = 1;

    zero_kernel<<<(N_NODES * 160 + 255) / 256, 256, 0, stream>>>(agg, stN, stE);
    pack_weights_kernel<<<64, 256, 0, stream>>>(WN, packN);
    pack_weights_kernel<<<64, 256, 0, stream>>>(WE, packE);
    node_pre_kernel<<<N_NODES, 160, 0, stream>>>(node_fea, n_pre0, n_pre1, nf);

    // ---- node-update branch ----
    mega_edge_kernel<<<N_EDGES / 16, 32, 0, stream>>>(
        nf, edge_fea, edge_sh, len_emb, idxI, idxJ, packN, n_b1, n_b2, n_b3, zbuf);
    scatter_add_kernel<<<N_EDGES, 160, 0, stream>>>(zbuf, idxI, agg);
    node_post_kernel<<<N_NODES, 160, 0, stream>>>(agg, node_fea, node_oh, n_post0,
                                                  n_post1, n_sc0, n_sc1, nf2);
    ln_stats1_kernel<<<(N_NODES + 255) / 256, 256, 0, stream>>>(nf2, batch, nullptr,
                                                                N_NODES, stN);
    ln_stats2_kernel<<<(N_NODES + 255) / 256, 256, 0, stream>>>(nf2, batch, nullptr,
                                                                N_NODES, stN);
    ln_finish_kernel<<<N_NODES, 160, 0, stream>>>(nf2, node_fea, batch, nullptr,
                                                  n_gamma, n_beta, stN, nodeOut);

    // ---- edge-update branch ----
    mega_edge_kernel<<<N_EDGES / 16, 32, 0, stream>>>(
        nodeOut, edge_fea, edge_sh, len_emb, idxI, idxJ, packE, e_b1, e_b2, e_b3, zbuf);
    edge_post_kernel<<<N_EDGES, 160, 0, stream>>>(zbuf, edge_fea, edge_oh, e_post0,
                                                  e_post1, e_sc0, e_sc1);
    ln_stats1_kernel<<<(N_EDGES + 255) / 256, 256, 0, stream>>>(zbuf, batch, idxI,
                                                                N_EDGES, stE);
    ln_stats2_kernel<<<(N_EDGES + 255) / 256, 256, 0, stream>>>(zbuf, batch, idxI,
                                                                N_EDGES, stE);
    ln_finish_kernel<<<N_EDGES, 160, 0, stream>>>(zbuf, edge_fea, batch, idxI,
                                                  e_gamma, e_beta, stE, (float*)d_out);
}